// AKConv_601295422149
// MI455X (gfx1250) — compile-verified
//
#include <hip/hip_runtime.h>
#include <hip/hip_bf16.h>

typedef float v2f __attribute__((ext_vector_type(2)));
typedef float v8f __attribute__((ext_vector_type(8)));

#define DIMS 64

// ---------------------------------------------------------------------------
// Kernel 1: zero the workspace (nbr accumulator [N*D] + degree [N]).
// Graph-replay safe: re-zeroes every launch.
// ---------------------------------------------------------------------------
__global__ void akconv_zero_kernel(float4* __restrict__ ws, int n4) {
    int i = blockIdx.x * blockDim.x + threadIdx.x;
    if (i < n4) ws[i] = make_float4(0.f, 0.f, 0.f, 0.f);
}

// ---------------------------------------------------------------------------
// Kernel 2: one wave32 per edge. Each lane gathers 2 consecutive floats of
// x[col] (global_load_b64, the 32 lanes cover the whole 256B row) and
// atomically adds them into nbr[row]. Lane 0 bumps the degree counter.
// ---------------------------------------------------------------------------
__global__ void akconv_edge_scatter_kernel(const float* __restrict__ x,
                                           const int* __restrict__ edge_index,
                                           float* __restrict__ nbr,
                                           float* __restrict__ deg,
                                           int E_) {
    long tid = (long)blockIdx.x * blockDim.x + threadIdx.x;
    int e    = (int)(tid >> 5);
    int lane = (int)(tid & 31);
    if (e >= E_) return;

    int row = edge_index[e];        // edge_index[0, e]
    int col = edge_index[E_ + e];   // edge_index[1, e]

    const float2 v = *(const float2*)(x + (long)col * DIMS + 2 * lane);
    float* dst = nbr + (long)row * DIMS + 2 * lane;
    atomicAdd(dst + 0, v.x);
    atomicAdd(dst + 1, v.y);
    if (lane == 0) atomicAdd(deg + row, 1.0f);
}

// ---------------------------------------------------------------------------
// Kernel 3: finalize. One wave per 16-node tile. Computes
//   agg = v1*x + v2*nbr     (VALU, fused into B-matrix construction)
//   out = diag(1/rowsum) * agg   via 4x V_WMMA_F32_16X16X4_F32 per 16x16 tile
// A-matrix 16x4 f32 layout (ISA 7.12.2): lane m in [0,16) holds row m;
//   VGPR0/1 = K {0,1} for lanes 0-15, K {2,3} for lanes 16-31.
// C/D 16x16 f32 layout: VGPR r holds (M = r + 8*laneHalf, N = lane&15).
// ---------------------------------------------------------------------------
__global__ void akconv_finalize_kernel(const float* __restrict__ x,
                                       const float* __restrict__ lambda_,
                                       const float* __restrict__ nbr,
                                       const float* __restrict__ deg,
                                       float* __restrict__ out,
                                       int N_) {
    const int waveInBlock = threadIdx.x >> 5;
    const int lane        = threadIdx.x & 31;
    const int tile        = blockIdx.x * (blockDim.x >> 5) + waveInBlock;
    const int m0          = tile * 16;
    if (m0 >= N_) return;          // uniform per wave -> EXEC all-1s for WMMA

    // lam = 1 + relu(lambda); v1 = (2lam-2)/lam; v2 = 2/lam
    const float lr  = lambda_[0];
    const float lam = 1.0f + (lr > 0.0f ? lr : 0.0f);
    const float v1  = (2.0f * lam - 2.0f) / lam;
    const float v2  = 2.0f / lam;

    const int m = lane & 15;   // A-matrix row / C-matrix column this lane owns
    const int g = lane >> 4;   // lane half selects K-pair (A/B) and M-half (C)

    // per-row inverse normalizer for node m0+m
    const float s = 1.0f / (v1 + v2 * deg[m0 + m]);

    for (int nt = 0; nt < 4; ++nt) {           // 4 column tiles cover D=64
        const int n0 = nt * 16;
        v8f c = {};
        #pragma unroll
        for (int ch = 0; ch < 4; ++ch) {       // diag(s) = sum of 4 16x4 slabs
            const int k0 = 4 * ch + 2 * g;     // K indices {k0, k0+1} this lane
            v2f a, b;
            a.x = (m == k0)     ? s : 0.0f;    // A(m, k0)
            a.y = (m == k0 + 1) ? s : 0.0f;    // A(m, k0+1)
            // B(k, n) = agg value of node (m0+k), feature (n0 + m)
            const int i0 = (m0 + k0) * DIMS + n0 + m;
            b.x = v1 * x[i0]        + v2 * nbr[i0];
            b.y = v1 * x[i0 + DIMS] + v2 * nbr[i0 + DIMS];
            c = __builtin_amdgcn_wmma_f32_16x16x4_f32(
                    /*neg_a=*/false, a, /*neg_b=*/false, b,
                    /*c_mod=*/(short)0, c, /*reuse_a=*/false, /*reuse_b=*/false);
        }
        // scatter C: VGPR r -> node (m0 + r + 8*g), feature (n0 + m)
        #pragma unroll
        for (int r = 0; r < 8; ++r) {
            out[(m0 + r + 8 * g) * DIMS + n0 + m] = c[r];
        }
    }
}

// ---------------------------------------------------------------------------
extern "C" void kernel_launch(void* const* d_in, const int* in_sizes, int n_in,
                              void* d_out, int out_size, void* d_ws, size_t ws_size,
                              hipStream_t stream) {
    const float* x          = (const float*)d_in[0];
    const float* lambda_    = (const float*)d_in[1];
    const int*   edge_index = (const int*)d_in[2];
    float*       out        = (float*)d_out;

    const int N_ = in_sizes[0] / DIMS;   // 100000
    const int E_ = in_sizes[2] / 2;      // 1600000

    float* nbr = (float*)d_ws;                    // N*D floats
    float* deg = nbr + (size_t)N_ * DIMS;         // N floats

    // 1) zero accumulators
    const int n4 = (N_ * DIMS + N_ + 3) / 4;
    akconv_zero_kernel<<<(n4 + 255) / 256, 256, 0, stream>>>((float4*)d_ws, n4);

    // 2) edge scatter: one wave32 per edge
    const long totalT = (long)E_ * 32;
    const int  eblocks = (int)((totalT + 255) / 256);
    akconv_edge_scatter_kernel<<<eblocks, 256, 0, stream>>>(x, edge_index, nbr, deg, E_);

    // 3) finalize with WMMA diag-normalization: 8 waves/block, 1 tile/wave
    const int tiles   = (N_ + 15) / 16;
    const int fblocks = (tiles + 7) / 8;
    akconv_finalize_kernel<<<fblocks, 256, 0, stream>>>(x, lambda_, nbr, deg, out, N_);
}